// ImputedMultiTimeAttentionV1_45148696215774
// MI455X (gfx1250) — compile-verified
//
#include <hip/hip_runtime.h>
#include <math.h>

// ImputedMultiTimeAttention for MI455X (gfx1250, wave32, WMMA + async LDS).
// B=16, TQ=TK=256, D=256, H=8, DK=32, F=36, SF=6.
// Masked per-feature softmax factorized as num=E@(mask*value), den=E@mask,
// with E = exp(S - rowmax). All GEMMs run on V_WMMA_F32_16X16X4_F32 to keep
// full fp32 precision (problem is memory-bound: ~4.2 GFLOP over ~50MB; the
// 34MB score scratch is L2-resident on the 192MB L2).
// attn_kernel stages the shared [mask*value | mask] operand into LDS with
// GLOBAL_LOAD_ASYNC_TO_LDS_B128 (ASYNCcnt) — the gfx1250 async copy path.

typedef __attribute__((ext_vector_type(2))) float v2f;
typedef __attribute__((ext_vector_type(8))) float v8f;

#define Bn  16
#define TQn 256
#define TKn 256
#define Dn  256
#define Fn  36
#define Hn  8
#define DKn 32
#define SFn 6
#define W2C 80   // padded: [0,36) num-src (mask*value) | [36,72) den-src (mask) | [72,80) zero pad

__device__ __forceinline__ v8f wmma4(v2f a, v2f b, v8f c) {
  // (neg_a, A, neg_b, B, c_mod, C, reuse_a, reuse_b)
  return __builtin_amdgcn_wmma_f32_16x16x4_f32(false, a, false, b, (short)0, c, false, false);
}

// gfx1250 async global->LDS copy (VGLOBAL encoding, tracked by ASYNCcnt).
// vdst VGPR holds the LDS byte address; flat "generic" pointers to __shared__
// carry the LDS offset in their low 32 bits (ISA 10.2 aperture mapping).
__device__ __forceinline__ void g2l_async_b128(void* lds, const void* gaddr) {
  unsigned loff = (unsigned)(size_t)lds;
  asm volatile("global_load_async_to_lds_b128 %0, %1, off"
               :: "v"(loff), "v"(gaddr) : "memory");
}
__device__ __forceinline__ void wait_async0() {
  asm volatile("s_wait_asynccnt 0x0" ::: "memory");
}

// ---------------------------------------------------------------- prep: W2
__global__ void prep_w2_kernel(const float* __restrict__ value,
                               const float* __restrict__ mask,
                               float* __restrict__ w2) {
  int b = blockIdx.x, k = threadIdx.x;                 // k in [0,256)
  const float* vp = value + (size_t)(b * TKn + k) * Fn;
  const float* mp = mask  + (size_t)(b * TKn + k) * Fn;
  float* wr = w2 + (size_t)(b * TKn + k) * W2C;
  #pragma unroll
  for (int f = 0; f < Fn; ++f) {
    float m = mp[f];
    wr[f]      = m * vp[f];
    wr[Fn + f] = m;
  }
  #pragma unroll
  for (int j = 2 * Fn; j < W2C; ++j) wr[j] = 0.f;
}

// ------------------------------------------------- proj: Q_all / K_all GEMM
// Out(4096x256) = X(4096x256) @ Wr(256x256), Wr[d][h*32+dk] = W[h][d][dk].
__global__ void proj_kernel(const float* __restrict__ query,
                            const float* __restrict__ key,
                            const float* __restrict__ Wq,
                            const float* __restrict__ Wk,
                            float* __restrict__ Qall,
                            float* __restrict__ Kall) {
  const int lane = threadIdx.x & 31;
  const int wave = threadIdx.x >> 5;
  const int lrow = lane & 15;
  const int lhi  = lane >> 4;
  const float* X = blockIdx.y ? key  : query;
  const float* W = blockIdx.y ? Wk   : Wq;
  float*       O = blockIdx.y ? Kall : Qall;

  int tile = blockIdx.x * 8 + wave;        // 4096 tiles: 256 m-tiles x 16 n-tiles
  int m0 = (tile & 255) * 16;
  int n0 = (tile >> 8) * 16;
  int n  = n0 + lrow;
  const float* Wcol = W + (size_t)(n >> 5) * (Dn * DKn) + (n & 31);
  const float* Xrow = X + (size_t)(m0 + lrow) * Dn;

  v8f c = {};
  #pragma unroll 4
  for (int kc = 0; kc < Dn; kc += 4) {
    int ka = kc + lhi * 2;                                   // A: K = lhi*2 + {0,1}
    v2f a = *(const v2f*)(Xrow + ka);
    v2f b; b[0] = Wcol[(size_t)ka * DKn]; b[1] = Wcol[(size_t)(ka + 1) * DKn];
    c = wmma4(a, b, c);
  }
  #pragma unroll
  for (int i = 0; i < 8; ++i)
    O[(size_t)(m0 + i + lhi * 8) * Dn + n] = c[i];
}

// --------------------------------------------------- scores: S = Q @ K^T
__global__ void scores_kernel(const float* __restrict__ Qall,
                              const float* __restrict__ Kall,
                              float* __restrict__ S) {
  const int lane = threadIdx.x & 31, wave = threadIdx.x >> 5;
  const int lrow = lane & 15, lhi = lane >> 4;
  int gt = blockIdx.x * 8 + wave;          // 32768 tiles = 128 (b,h) x 256 tiles
  int bh = gt >> 8;
  int t  = gt & 255;
  int b = bh >> 3, h = bh & 7;
  int m0 = (t & 15) * 16;
  int n0 = (t >> 4) * 16;
  const float* Arow = Qall + ((size_t)b * TQn + m0 + lrow) * Dn + h * DKn;
  const float* Brow = Kall + ((size_t)b * TKn + n0 + lrow) * Dn + h * DKn;  // K^T column = K row

  v8f c = {};
  #pragma unroll
  for (int kc = 0; kc < DKn; kc += 4) {
    int ka = kc + lhi * 2;
    v2f a  = *(const v2f*)(Arow + ka);
    v2f bb = *(const v2f*)(Brow + ka);
    c = wmma4(a, bb, c);
  }
  float* Sb = S + (size_t)bh * TQn * TKn;
  #pragma unroll
  for (int i = 0; i < 8; ++i)
    Sb[(size_t)(m0 + i + lhi * 8) * TKn + n0 + lrow] = c[i];
}

// -------------------------------------------------------- rowmax per score row
__global__ void rowmax_kernel(const float* __restrict__ S, float* __restrict__ rmax) {
  const int lane = threadIdx.x & 31, wave = threadIdx.x >> 5;
  int row = blockIdx.x * 8 + wave;         // 32768 rows
  const float* sr = S + (size_t)row * TKn;
  float m = sr[lane];
  #pragma unroll
  for (int j = 1; j < 8; ++j) m = fmaxf(m, sr[lane + j * 32]);
  #pragma unroll
  for (int off = 16; off > 0; off >>= 1) m = fmaxf(m, __shfl_xor(m, off, 32));
  if (lane == 0) rmax[row] = m;
}

// --------------------------- attn: [num|den] = exp(S - rmax) @ W2, head = num/den
// One block = 8 waves, all sharing (b,h); each wave owns one 16-row q-tile.
// W2(b) is consumed in four 64-row k-panels staged into LDS via async copies.
#define KPAN 64
__global__ void attn_kernel(const float* __restrict__ S,
                            const float* __restrict__ rmax,
                            const float* __restrict__ w2,
                            float* __restrict__ Hd) {
  // 40KB: k-panel (64x80 = 5120 floats) aliases the post-loop num/den tiles
  // (8 waves x 16x80 = 10240 floats); lifetimes are disjoint.
  __shared__ __align__(16) float smem[10240];
  const int tid  = threadIdx.x;
  const int lane = tid & 31, wave = tid >> 5;
  const int lrow = lane & 15, lhi = lane >> 4;
  int bh   = blockIdx.x >> 1;              // 128 (b,h) pairs, 2 blocks each
  int half = blockIdx.x & 1;
  int qt   = half * 8 + wave;              // q-tile 0..15
  int b = bh >> 3, h = bh & 7;
  int m0 = qt * 16;
  const float* Srow = S + ((size_t)bh * TQn + m0 + lrow) * TKn;
  float rm = rmax[(size_t)bh * TQn + m0 + lrow];
  const float* W2b = w2 + (size_t)b * TKn * W2C;

  v8f c0 = {}, c1 = {}, c2 = {}, c3 = {}, c4 = {};
  for (int k0 = 0; k0 < TKn; k0 += KPAN) {
    // Stage 64x80 W2 panel: 1280 float4 -> 5 async b128 copies per thread.
    #pragma unroll
    for (int j = 0; j < 5; ++j) {
      int idx = tid + j * 256;             // float4 index in panel
      g2l_async_b128(&smem[(size_t)idx * 4], W2b + (size_t)k0 * W2C + (size_t)idx * 4);
    }
    wait_async0();                          // own ASYNCcnt drained
    __syncthreads();                        // panel visible to all waves

    #pragma unroll 4
    for (int kc = 0; kc < KPAN; kc += 4) {
      int ka = kc + lhi * 2;
      v2f sv = *(const v2f*)(Srow + k0 + ka);
      v2f a; a[0] = __expf(sv[0] - rm); a[1] = __expf(sv[1] - rm); // each E elem once
      const float* pw0 = &smem[(size_t)ka * W2C + lrow];
      const float* pw1 = pw0 + W2C;
      v2f b0; b0[0] = pw0[0];  b0[1] = pw1[0];
      v2f b1; b1[0] = pw0[16]; b1[1] = pw1[16];
      v2f b2; b2[0] = pw0[32]; b2[1] = pw1[32];
      v2f b3; b3[0] = pw0[48]; b3[1] = pw1[48];
      v2f b4; b4[0] = pw0[64]; b4[1] = pw1[64];
      c0 = wmma4(a, b0, c0);
      c1 = wmma4(a, b1, c1);
      c2 = wmma4(a, b2, c2);
      c3 = wmma4(a, b3, c3);
      c4 = wmma4(a, b4, c4);
    }
    __syncthreads();                        // panel consumed before overwrite
  }

  // Spill num|den tile to (wave-private) LDS, then divide and scatter.
  float* acc = &smem[(size_t)wave * 16 * W2C];
  #pragma unroll
  for (int i = 0; i < 8; ++i) {
    int r = i + lhi * 8;
    acc[r * W2C +  0 + lrow] = c0[i];
    acc[r * W2C + 16 + lrow] = c1[i];
    acc[r * W2C + 32 + lrow] = c2[i];
    acc[r * W2C + 48 + lrow] = c3[i];
    acc[r * W2C + 64 + lrow] = c4[i];
  }
  for (int idx = lane; idx < 16 * Fn; idx += 32) {
    int r = idx / Fn, f = idx - r * Fn;
    float num = acc[r * W2C + f];
    float den = acc[r * W2C + Fn + f];
    float hv = (den > 0.f) ? (num / den) : 0.f;   // all-masked column ~ prob 2^-256
    Hd[(((size_t)b * TQn + m0 + r) * Fn + f) * Hn + h] = hv;
  }
}

// ------------------------------- tail: tanh(head@Wc) -> tanh(@Wo1) -> @Wo2
__global__ void tail_kernel(const float* __restrict__ Hd,
                            const float* __restrict__ Wc,
                            const float* __restrict__ bc,
                            const float* __restrict__ Wo1,
                            const float* __restrict__ bo1,
                            const float* __restrict__ Wo2,
                            const float* __restrict__ bo2,
                            float* __restrict__ out) {
  __shared__ float hh[Fn * Hn];    // 288: head[f][h] for this (b,q)
  __shared__ float w1[Fn * SFn];   // 216
  __shared__ float w2s[SFn];
  __shared__ float b1s[SFn];
  int bq  = blockIdx.x;            // 0..4095 = b*256+q
  int tid = threadIdx.x;           // d
  const float* hp = Hd + (size_t)bq * (Fn * Hn);
  for (int i = tid; i < Fn * Hn; i += 256) hh[i] = hp[i];
  for (int i = tid; i < Fn * SFn; i += 256) w1[i] = Wo1[i];
  if (tid < SFn) { w2s[tid] = Wo2[tid]; b1s[tid] = bo1[tid]; }
  __syncthreads();

  float wc[Hn];
  #pragma unroll
  for (int h = 0; h < Hn; ++h) wc[h] = Wc[h * Dn + tid];
  float bcd = bc[tid];
  float ts[SFn] = {0.f, 0.f, 0.f, 0.f, 0.f, 0.f};
  for (int f = 0; f < Fn; ++f) {
    float a = bcd;
    #pragma unroll
    for (int h = 0; h < Hn; ++h) a = fmaf(hh[f * Hn + h], wc[h], a);
    float lat = tanhf(a);
    #pragma unroll
    for (int s = 0; s < SFn; ++s) ts[s] = fmaf(lat, w1[f * SFn + s], ts[s]);
  }
  float o = bo2[tid];
  #pragma unroll
  for (int s = 0; s < SFn; ++s) o = fmaf(tanhf(ts[s] + b1s[s]), w2s[s], o);
  out[(size_t)bq * Dn + tid] = o;
}

// ---------------------------------------------------------------- launcher
extern "C" void kernel_launch(void* const* d_in, const int* in_sizes, int n_in,
                              void* d_out, int out_size, void* d_ws, size_t ws_size,
                              hipStream_t stream) {
  const float* query = (const float*)d_in[0];
  const float* key   = (const float*)d_in[1];
  const float* value = (const float*)d_in[2];
  const float* mask  = (const float*)d_in[3];
  const float* Wq    = (const float*)d_in[4];
  const float* Wk    = (const float*)d_in[5];
  const float* Wc    = (const float*)d_in[6];
  const float* bc    = (const float*)d_in[7];
  const float* Wo1   = (const float*)d_in[8];
  const float* bo1   = (const float*)d_in[9];
  const float* Wo2   = (const float*)d_in[10];
  const float* bo2   = (const float*)d_in[11];
  float* out = (float*)d_out;

  // Scratch layout (floats). Total = 12,025,856 floats = ~48.1 MB.
  float* ws   = (float*)d_ws;
  float* Qall = ws;                    // 4096*256   = 1,048,576
  float* Kall = Qall + 1048576;        //             1,048,576
  float* S    = Kall + 1048576;        // 128*256*256 = 8,388,608 (fits L2: 192MB)
  float* rmax = S + 8388608;           //                 32,768
  float* w2   = rmax + 32768;          // 16*256*80  =   327,680
  float* Hd   = w2 + 327680;           // 16*256*36*8 = 1,179,648

  prep_w2_kernel<<<dim3(Bn),      256, 0, stream>>>(value, mask, w2);
  proj_kernel   <<<dim3(512, 2),  256, 0, stream>>>(query, key, Wq, Wk, Qall, Kall);
  scores_kernel <<<dim3(4096),    256, 0, stream>>>(Qall, Kall, S);
  rowmax_kernel <<<dim3(4096),    256, 0, stream>>>(S, rmax);
  attn_kernel   <<<dim3(256),     256, 0, stream>>>(S, rmax, w2, Hd);
  tail_kernel   <<<dim3(4096),    256, 0, stream>>>(Hd, Wc, bc, Wo1, bo1, Wo2, bo2, out);

  (void)in_sizes; (void)n_in; (void)out_size; (void)ws_size;
}